// GuidedChannelWiseConv2d_75007308857581
// MI455X (gfx1250) — compile-verified
//
#include <hip/hip_runtime.h>
#include <stdint.h>

// Problem constants from the reference: B=8, Ci=64, H=W=128, K=3x3, SAME pad.
#define B_      8
#define CI      64
#define HH      128
#define WW      128
#define KK      9
#define TH      8               // output rows per block tile
#define BLOCK   256             // 8 waves (wave32); wave i == tile row i
#define TSTRIDE 136             // floats; row byte-stride 544 % 16 == 0
#define TROWS   (TH + 2)        // +row halo
#define TELEMS  (TROWS * TSTRIDE)   // 1360 floats = 5440 B LDS
// tile layout per row: [0..2] unused pad, [3] gw=-1 (always 0),
// [4..131] gw=0..127 (16B-aligned interior), [132] gw=128 (always 0), [133..135] pad

typedef float v4f __attribute__((ext_vector_type(4)));

__global__ __launch_bounds__(BLOCK)
void guided_cw_conv2d_gfx1250(const float* __restrict__ inp,   // (B*Ci, H, W)
                              const float* __restrict__ wgt,   // (B*Ci, 9, H, W)
                              float* __restrict__ out)         // (B*Ci, H, W)
{
  __shared__ float tile[TELEMS];

  const int tid = threadIdx.x;
  const int bc  = blockIdx.x;           // plane (b*Ci + c)
  const int h0  = blockIdx.y * TH;      // first output row of this tile

  // ---- phase 0: zero the tile (vectorized); cells that stay zero provide
  //      the SAME-padding: col cells [3] and [132] always, OOB rows fully. ---
  {
    v4f z; z.x = 0.f; z.y = 0.f; z.z = 0.f; z.w = 0.f;
    const int i0 = tid * 4;
    if (i0 < TELEMS)        *(v4f*)&tile[i0] = z;           // 16B-aligned
    const int i1 = i0 + BLOCK * 4;
    if (i1 < TELEMS)        *(v4f*)&tile[i1] = z;
  }
  __syncthreads();   // zero-stores complete (DScnt) before async LDS writes

  // ---- phase 1: async global -> LDS, one B128 per wave-row ----------------
  // GVS addressing: saddr = 64-bit plane base, vaddr = per-lane i32 byte off.
  const uint64_t sbase = (uint64_t)(inp + (size_t)bc * (HH * WW));
  const unsigned lds0  = (unsigned)(size_t)(&tile[0]);  // low 32 bits == LDS offset
  const int lane = tid & 31;
  for (int tr = (tid >> 5); tr < TROWS; tr += 8) {      // 10 rows over 8 waves
    const int gh = h0 - 1 + tr;                         // tile row 0 == row h0-1
    if ((unsigned)gh < (unsigned)HH) {                  // OOB rows stay zero
      const unsigned ldsa = lds0 + (unsigned)(tr * TSTRIDE + 4) * 4u
                                 + (unsigned)lane * 16u;          // 16B aligned
      const int goff = gh * (WW * 4) + lane * 16;       // lane*16B, coalesced
      // copies 16 bytes from global address sbase plus goff to LDS address ldsa; tracked by ASYNCcnt (CDNA5)
      asm volatile("global_load_async_to_lds_b128 %0, %1, %2"
                   :
                   : "v"(ldsa), "v"(goff), "s"(sbase)
                   : "memory");
    }
  }
  asm volatile("s_wait_asynccnt 0x0" ::: "memory");  // own wave's copies done
  __syncthreads();                                   // all waves' LDS writes visible

  // ---- phase 2: compute 4 pixels/lane, 9 taps, branch-free ----------------
  const int r  = tid >> 5;            // wave id == tile output row 0..7
  const int c4 = lane << 2;           // cols c4..c4+3 (16B aligned in global)
  const int h  = h0 + r;

  const float* wbase = wgt + ((size_t)bc * KK) * (HH * WW) + (size_t)h * WW + c4;

  float a0 = 0.f, a1 = 0.f, a2 = 0.f, a3 = 0.f;
#pragma unroll
  for (int k = 0; k < KK; ++k) {
    const int dh = k / 3;
    const int dw = k - dh * 3;
    // weights: 302 MB streamed once, zero reuse -> non-temporal
    const v4f wk = __builtin_nontemporal_load(
        (const v4f*)(wbase + (size_t)k * (HH * WW)));
    // input (h-1+dh, c4+j-1+dw) lives at tile[(r+dh)*136 + 3 + c4 + j + dw]
    const float* trow = &tile[(r + dh) * TSTRIDE + 3 + c4 + dw];
    a0 += trow[0] * wk.x;
    a1 += trow[1] * wk.y;
    a2 += trow[2] * wk.z;
    a3 += trow[3] * wk.w;
  }

  v4f o; o.x = a0; o.y = a1; o.z = a2; o.w = a3;
  __builtin_nontemporal_store(
      o, (v4f*)(out + (size_t)bc * (HH * WW) + (size_t)h * WW + c4));
}

extern "C" void kernel_launch(void* const* d_in, const int* in_sizes, int n_in,
                              void* d_out, int out_size, void* d_ws, size_t ws_size,
                              hipStream_t stream) {
  const float* inp = (const float*)d_in[0];   // (8,64,128,128) f32
  const float* wgt = (const float*)d_in[1];   // (8,64,9,128,128) f32
  float* out = (float*)d_out;                 // (8,64,128,128) f32

  dim3 grid(B_ * CI, HH / TH);                // 512 x 16 blocks
  guided_cw_conv2d_gfx1250<<<grid, BLOCK, 0, stream>>>(inp, wgt, out);
}